// AgentX_42116449304641
// MI455X (gfx1250) — compile-verified
//
#include <hip/hip_runtime.h>
#include <math.h>

// ---------------------------------------------------------------------------
// ConvLSTM2D (B=8,T=16,H=W=256,C=3, F=36,K=5, stride 4 valid input conv,
// stride 1 SAME recurrent conv) + flatten, on gfx1250 via fp32 WMMA 16x16x4.
// Recurrent GEMM uses channel padding 36->40 + K re-permutation so every A
// fragment is one aligned b128 covering two WMMA chunks; B frags are b128 DS
// loads.  Depth-3 software pipeline + sched_group_barrier interleaving.
// ---------------------------------------------------------------------------

typedef __attribute__((ext_vector_type(2))) float v2f;
typedef __attribute__((ext_vector_type(4))) float v4f;
typedef __attribute__((ext_vector_type(8))) float v8f;

#define HO     63
#define WO     63
#define SPAT   (HO * WO)           // 3969
#define FDIM   36
#define FP     40                  // padded channel dim in hp (36 real + 4 zero)
#define GDIM   144                 // 4*F
#define BATCH  8
#define TSTEPS 16
#define HIN    256
#define CIN    3
#define KROW   15                  // 5*3 contiguous floats per kernel row
#define KAPP   80                  // xconv: 5 rows x 16 (15 real + 1 zero pad)
#define KBP    1000                // recgemm: 5*5*40 padded K
#define MA     (BATCH * TSTEPS * SPAT)   // 508032
#define MB     (BATCH * SPAT)            // 31752
#define HPAD   67                  // 63 + 2*2 SAME halo
#define HPROW  (HPAD * FP)         // 2680 floats per padded-h row

#if __has_builtin(__builtin_amdgcn_sched_group_barrier)
#define SCHED_GROUPS(nvm, nds, nmm)                                \
  do {                                                             \
    __builtin_amdgcn_sched_group_barrier(0x020, (nvm), 0); /*VMEM rd*/ \
    __builtin_amdgcn_sched_group_barrier(0x100, (nds), 0); /*DS rd*/   \
    __builtin_amdgcn_sched_group_barrier(0x008, (nmm), 0); /*WMMA*/    \
  } while (0)
#else
#define SCHED_GROUPS(nvm, nds, nmm) do {} while (0)
#endif

// D = A(16x4) * B(4x16) + C, exact fp32 on the matrix pipe.
static __device__ __forceinline__ v8f wmma_f32(v2f a, v2f b, v8f c) {
  return __builtin_amdgcn_wmma_f32_16x16x4_f32(
      /*neg_a=*/false, a, /*neg_b=*/false, b,
      /*c_mod=*/(short)0, c, /*reuse_a=*/false, /*reuse_b=*/false);
}

// ---------------------------------------------------------------------------
// Zero hp (padded h, incl. channel padding) + c state.
// ---------------------------------------------------------------------------
__global__ __launch_bounds__(256)
void AgentX_zero(float* __restrict__ p, int n) {
  int i = blockIdx.x * 256 + threadIdx.x;
  if (i < n) p[i] = 0.0f;
}

// ---------------------------------------------------------------------------
// Phase A: xconv[m][n] = bias[n] + sum_k patch(m,k) * Wx[k][n]
//   K re-permuted to 5 rows x 16 (slot 15 per row is a zero weight); every A
//   fragment is a contiguous in-bounds b64 at an immediate offset.
// ---------------------------------------------------------------------------
__global__ __launch_bounds__(256) __attribute__((amdgpu_waves_per_eu(1)))
void AgentX_xconv(const float* __restrict__ x, const float* __restrict__ Wx,
                  const float* __restrict__ bias, float* __restrict__ xconv) {
  __shared__ float wlds[KAPP * 16];                // pair-swizzled B slice
  const int n0 = blockIdx.y * 16;

  for (int idx = threadIdx.x; idx < KAPP * 16; idx += 256) {
    int j = idx & 15, kp = idx >> 4;               // kp in [0,80)
    int ky = kp >> 4, r = kp & 15;                 // r==15 -> zero pad column
    float v = (r < KROW) ? Wx[(ky * KROW + r) * GDIM + n0 + j] : 0.0f;
    wlds[(((kp >> 1) * 16 + j) << 1) + (kp & 1)] = v;
  }
  __syncthreads();

  const int lane = threadIdx.x & 31;
  const int wave = threadIdx.x >> 5;
  const int col  = lane & 15;        // N column / A row
  const int hf   = lane >> 4;        // K half-select
  const int mBase = (blockIdx.x * 32 + wave * 4) * 16;

  const float* xb[4];
  v8f acc[4];
  const float bv = bias[n0 + col];
  #pragma unroll
  for (int i = 0; i < 4; ++i) {
    int m = mBase + i * 16 + col;
    if (m >= MA) m = MA - 1;                       // clamp, store is guarded
    int ox = m % WO; int tmp = m / WO;
    int oy = tmp % HO; int bt = tmp / HO;
    xb[i] = x + ((bt * HIN + oy * 4) * HIN + ox * 4) * CIN + 2 * hf;
    acc[i] = (v8f){bv, bv, bv, bv, bv, bv, bv, bv};
  }
  const float* lb = &wlds[(hf * 16 + col) * 2];    // per-lane LDS base

  v2f Ab[4][4]; v2f Bb[4];
#define LOADA(dst, c) do {                                                   \
    Bb[dst]    = *(const v2f*)(lb + (c) * 64);                               \
    Ab[dst][0] = *(const v2f*)(xb[0] + ((c) >> 2) * (HIN * CIN) + ((c) & 3) * 4); \
    Ab[dst][1] = *(const v2f*)(xb[1] + ((c) >> 2) * (HIN * CIN) + ((c) & 3) * 4); \
    Ab[dst][2] = *(const v2f*)(xb[2] + ((c) >> 2) * (HIN * CIN) + ((c) & 3) * 4); \
    Ab[dst][3] = *(const v2f*)(xb[3] + ((c) >> 2) * (HIN * CIN) + ((c) & 3) * 4); \
  } while (0)

  #pragma unroll
  for (int c = 0; c < 20; ++c) {
    if (c == 0) { LOADA(0, 0); LOADA(1, 1); LOADA(2, 2); LOADA(3, 3); }
    const int s = c & 3;
    acc[0] = wmma_f32(Ab[s][0], Bb[s], acc[0]);
    acc[1] = wmma_f32(Ab[s][1], Bb[s], acc[1]);
    acc[2] = wmma_f32(Ab[s][2], Bb[s], acc[2]);
    acc[3] = wmma_f32(Ab[s][3], Bb[s], acc[3]);
    if (c + 4 < 20) LOADA(s, c + 4);
    SCHED_GROUPS(4, 1, 4);
  }
#undef LOADA

  #pragma unroll
  for (int i = 0; i < 4; ++i)
    #pragma unroll
    for (int v = 0; v < 8; ++v) {
      int mrow = mBase + i * 16 + v + 8 * hf;      // C/D layout: M = v + 8*half
      if (mrow < MA) xconv[mrow * GDIM + n0 + col] = acc[i][v];
    }
}

// ---------------------------------------------------------------------------
// Phase B GEMM: z[m][n] = xconv[b,t,sp][n] + sum_k hp_patch(m,k) * Wh[k][n]
//   hp padded to 40 channels; K' = 5*5*40 = 1000, permuted in superchunks of
//   8: lanes 0-15 take K 8s..8s+3, lanes 16-31 take 8s+4..8s+7 -> one aligned
//   b128 per lane per superchunk covers two WMMA chunks (.xy / .zw).  Wh
//   slice staged in LDS with the same permutation (64 KB), read as b128.
// ---------------------------------------------------------------------------
__global__ __launch_bounds__(256) __attribute__((amdgpu_waves_per_eu(1)))
void AgentX_recgemm(const float* __restrict__ hp, const float* __restrict__ Wh,
                    const float* __restrict__ xconv, float* __restrict__ z,
                    int t) {
  __shared__ float wlds[KBP * 16];                 // 64,000 B
  const int n0 = blockIdx.y * 16;

  // slds[(k>>2)*64 + j*4 + (k&3)] = Wh'[k][n0+j], Wh' zero for f in [36,40)
  for (int idx = threadIdx.x; idx < KBP * 16; idx += 256) {
    int j = idx & 15, k = idx >> 4;                // k in [0,1000)
    int kk = k / FP, f = k - kk * FP;              // kk = ky*5+kx
    float v = (f < FDIM) ? Wh[(kk * FDIM + f) * GDIM + n0 + j] : 0.0f;
    wlds[(k >> 2) * 64 + j * 4 + (k & 3)] = v;
  }
  __syncthreads();

  const int lane = threadIdx.x & 31;
  const int wave = threadIdx.x >> 5;
  const int col  = lane & 15;
  const int hf   = lane >> 4;
  const int mBase = (blockIdx.x * 32 + wave * 4) * 16;

  const float* hb[4];
  v8f acc[4];
  #pragma unroll
  for (int i = 0; i < 4; ++i) {
    int m = mBase + i * 16 + col;
    if (m >= MB) m = MB - 1;
    int ox = m % WO; int tmp = m / WO;
    int oy = tmp % HO; int b = tmp / HO;
    // top-left of 5x5 window in padded h; *FP and 4*hf keep 16B alignment
    hb[i] = hp + ((b * HPAD + oy) * HPAD + ox) * FP + 4 * hf;
    #pragma unroll
    for (int v = 0; v < 8; ++v) {
      int mrow = mBase + i * 16 + v + 8 * hf;
      if (mrow >= MB) mrow = MB - 1;
      int sp = mrow % SPAT; int bb = mrow / SPAT;
      acc[i][v] = xconv[((bb * TSTEPS + t) * SPAT + sp) * GDIM + n0 + col];
    }
  }
  const float* lb = &wlds[hf * 64 + col * 4];      // per-lane LDS base

  v4f A4[3][4]; v4f B4[3];
  for (int ky = 0; ky < 5; ++ky) {                 // runtime loop: ptr bumps
    const float* hr0 = hb[0] + ky * HPROW;
    const float* hr1 = hb[1] + ky * HPROW;
    const float* hr2 = hb[2] + ky * HPROW;
    const float* hr3 = hb[3] + ky * HPROW;
    const float* lr  = lb + ky * (25 * 128);       // 25 superchunks per ky

    // superchunk s in [0,25): kx=s/5, sf=s%5; offsets fold to immediates.
#define LOADS(dst, s) do {                                                   \
    B4[dst]    = *(const v4f*)(lr + (s) * 128);                              \
    A4[dst][0] = *(const v4f*)(hr0 + ((s) / 5) * FP + ((s) % 5) * 8);        \
    A4[dst][1] = *(const v4f*)(hr1 + ((s) / 5) * FP + ((s) % 5) * 8);        \
    A4[dst][2] = *(const v4f*)(hr2 + ((s) / 5) * FP + ((s) % 5) * 8);        \
    A4[dst][3] = *(const v4f*)(hr3 + ((s) / 5) * FP + ((s) % 5) * 8);        \
  } while (0)

    #pragma unroll
    for (int s = 0; s < 25; ++s) {
      if (s == 0) { LOADS(0, 0); LOADS(1, 1); LOADS(2, 2); }
      const int sl = s % 3;
      acc[0] = wmma_f32(A4[sl][0].xy, B4[sl].xy, acc[0]);
      acc[1] = wmma_f32(A4[sl][1].xy, B4[sl].xy, acc[1]);
      acc[2] = wmma_f32(A4[sl][2].xy, B4[sl].xy, acc[2]);
      acc[3] = wmma_f32(A4[sl][3].xy, B4[sl].xy, acc[3]);
      acc[0] = wmma_f32(A4[sl][0].zw, B4[sl].zw, acc[0]);
      acc[1] = wmma_f32(A4[sl][1].zw, B4[sl].zw, acc[1]);
      acc[2] = wmma_f32(A4[sl][2].zw, B4[sl].zw, acc[2]);
      acc[3] = wmma_f32(A4[sl][3].zw, B4[sl].zw, acc[3]);
      if (s + 3 < 25) LOADS(sl, s + 3);
      SCHED_GROUPS(4, 1, 8);
    }
#undef LOADS
  }

  #pragma unroll
  for (int i = 0; i < 4; ++i)
    #pragma unroll
    for (int v = 0; v < 8; ++v) {
      int mrow = mBase + i * 16 + v + 8 * hf;
      if (mrow < MB) z[mrow * GDIM + n0 + col] = acc[i][v];
    }
}

// ---------------------------------------------------------------------------
// Gates: i,f,c,o split of z; hard_sigmoid = clip(0.2x+0.5, 0, 1).
// Updates cell state, writes h into hp interior (f<36; padding stays zero).
// ---------------------------------------------------------------------------
__global__ __launch_bounds__(256)
void AgentX_gate(const float* __restrict__ z, float* __restrict__ cbuf,
                 float* __restrict__ hp, float* __restrict__ out) {
  int idx = blockIdx.x * 256 + threadIdx.x;
  if (idx >= MB * FDIM) return;
  int f = idx % FDIM; int m = idx / FDIM;
  int ox = m % WO; int tmp = m / WO;
  int oy = tmp % HO; int b = tmp / HO;

  const float* zp = z + m * GDIM + f;
  float zi = zp[0], zf = zp[FDIM], zc = zp[2 * FDIM], zo = zp[3 * FDIM];
  float ig = fminf(fmaxf(0.2f * zi + 0.5f, 0.0f), 1.0f);
  float fg = fminf(fmaxf(0.2f * zf + 0.5f, 0.0f), 1.0f);
  float og = fminf(fmaxf(0.2f * zo + 0.5f, 0.0f), 1.0f);

  float cn = fg * cbuf[idx] + ig * tanhf(zc);
  float h  = og * tanhf(cn);
  cbuf[idx] = cn;
  hp[((b * HPAD + oy + 2) * HPAD + (ox + 2)) * FP + f] = h;
  out[idx] = h;                                     // (B, Ho*Wo*F) flatten
}

// ---------------------------------------------------------------------------
// Launch: zero -> xconv -> 16 x (recgemm, gate).  Workspace layout (floats):
//   xconv: 73,156,608 | hp(40ch): 1,436,480 | c: 1,143,072 | z: 4,572,288
//   total ~321.2 MB.
// ---------------------------------------------------------------------------
extern "C" void kernel_launch(void* const* d_in, const int* in_sizes, int n_in,
                              void* d_out, int out_size, void* d_ws,
                              size_t ws_size, hipStream_t stream) {
  const float* x  = (const float*)d_in[0];   // states_in (8,16,256,256,3)
  const float* Wx = (const float*)d_in[1];   // (5,5,3,144)
  const float* Wh = (const float*)d_in[2];   // (5,5,36,144)
  const float* b  = (const float*)d_in[3];   // (144,)
  float* out = (float*)d_out;

  float* ws    = (float*)d_ws;
  float* xconv = ws;
  float* hp    = xconv + (size_t)MA * GDIM;
  float* cbuf  = hp + (size_t)BATCH * HPAD * HPAD * FP;
  float* zbuf  = cbuf + (size_t)MB * FDIM;

  (void)in_sizes; (void)n_in; (void)out_size; (void)ws_size;

  // hp (incl. channel padding) and cbuf are contiguous -> one zero pass.
  const int nz = BATCH * HPAD * HPAD * FP + MB * FDIM;
  AgentX_zero<<<(nz + 255) / 256, 256, 0, stream>>>(hp, nz);

  dim3 gA((MA + 511) / 512, 9);                  // 8 waves x 4 M-tiles, 9 N-tiles
  AgentX_xconv<<<gA, 256, 0, stream>>>(x, Wx, b, xconv);

  dim3 gB((MB + 511) / 512, 9);
  const int gG = (MB * FDIM + 255) / 256;
  for (int t = 0; t < TSTEPS; ++t) {
    AgentX_recgemm<<<gB, 256, 0, stream>>>(hp, Wh, xconv, zbuf, t);
    AgentX_gate<<<gG, 256, 0, stream>>>(zbuf, cbuf, hp, out);
  }
}